// SO3Encode_83631603187719
// MI455X (gfx1250) — compile-verified
//
#include <hip/hip_runtime.h>
#include <hip/hip_bf16.h>

// SO3 encode: mean = Rodrigues rotation from skew(a,b,c); logvar = 5.4*sigmoid-9.2.
// Bandwidth-bound elementwise op: 302 MB traffic -> ~13us floor @ 23.3 TB/s.
// WMMA inapplicable: no reduction dim (A@A for skew A == w*w^T - |w|^2*I closed
// form), so the gfx1250 win is data movement:
//   * coalesced global_load_b96 inputs (12B/lane contiguous)
//   * branchless small-angle select (v_cndmask; no exec divergence)
//   * LDS-staged outputs dumped via GLOBAL_STORE_ASYNC_FROM_LDS_B128
//     (direct LDS->HBM on ASYNCcnt, no VGPR round trip)
//   * tail handling split into a separate kernel so the hot kernel has no
//     ragged-block branch (tail kernel never launches for B=262144)

#define NG   16
#define BLK  256

__device__ __forceinline__ void so3_item(float a, float b, float c,
                                         float l0, float l1, float l2,
                                         float* __restrict__ m,   // 9 floats
                                         float* __restrict__ lo)  // 3 floats
{
    const float th2 = a * a + b * b + c * c;

    // Branchless where(small, ...) exactly like the reference.
    const bool  small = th2 < 1e-8f;
    const float t2    = small ? 1.0f : th2;          // safe denominator
    const float t     = __builtin_sqrtf(t2);
    const float inv   = __builtin_amdgcn_rcpf(t2);   // v_rcp_f32
    float s1 = __sinf(t) * t * inv;                  // sin(t)/t
    float s2 = (1.0f - __cosf(t)) * inv;             // (1-cos(t))/t^2
    s1 = small ? 1.0f - th2 * (1.0f / 6.0f)  : s1;
    s2 = small ? 0.5f - th2 * (1.0f / 24.0f) : s2;

    // A = [[0,a,b],[-a,0,c],[-b,-c,0]] = [w]x, w = (-c,b,-a); A^2 = w w^T - th2*I
    const float aa = a * a, bb = b * b, cc = c * c;
    const float ab = a * b, ac = a * c, bc = b * c;
    const float s1a = s1 * a, s1b = s1 * b, s1c = s1 * c;

    m[0] = 1.0f - s2 * (aa + bb);
    m[1] =  s1a - s2 * bc;
    m[2] =  s1b + s2 * ac;
    m[3] = -s1a - s2 * bc;
    m[4] = 1.0f - s2 * (aa + cc);
    m[5] =  s1c - s2 * ab;
    m[6] = -s1b + s2 * ac;
    m[7] = -s1c - s2 * ab;
    m[8] = 1.0f - s2 * (bb + cc);

    lo[0] = fmaf(5.4f, __builtin_amdgcn_rcpf(1.0f + __expf(-l0)), -9.2f);
    lo[1] = fmaf(5.4f, __builtin_amdgcn_rcpf(1.0f + __expf(-l1)), -9.2f);
    lo[2] = fmaf(5.4f, __builtin_amdgcn_rcpf(1.0f + __expf(-l2)), -9.2f);
}

// Hot kernel: every block is full (gridDim = total/BLK blocks of BLK items).
__global__ __launch_bounds__(BLK)
void so3_encode_main(const float* __restrict__ z,
                     float* __restrict__ out_mean,   // [total, 9]
                     float* __restrict__ out_logvar) // [total, 3]
{
    __shared__ __align__(16) float sM[BLK * 9];  // 9216 B
    __shared__ __align__(16) float sL[BLK * 3];  // 3072 B

    const int tid   = threadIdx.x;
    const int item0 = blockIdx.x * BLK;
    const int item  = item0 + tid;

    {
        const int brow = item >> 4;
        const int g    = item & (NG - 1);
        const float* __restrict__ va = z + (size_t)brow * (6 * NG) + 3 * g;
        const float* __restrict__ vl = va + 3 * NG;
        // Coalesced: 12B/lane contiguous -> global_load_b96 x2.
        so3_item(va[0], va[1], va[2], vl[0], vl[1], vl[2],
                 sM + tid * 9 /* stride 9, coprime to 64 banks */,
                 sL + tid * 3);
    }

    __syncthreads();

    // Contiguous async LDS->global b128 dump; block regions are 16B aligned.
    const float4* sM4 = (const float4*)sM;  // 576 vec4
    const float4* sL4 = (const float4*)sL;  // 192 vec4
    float4* gM4 = (float4*)(out_mean   + (size_t)item0 * 9);
    float4* gL4 = (float4*)(out_logvar + (size_t)item0 * 3);

    #pragma unroll
    for (int k = 0; k < 3; ++k) {
        const int idx = k * BLK + tid;
        if (idx < (BLK * 9) / 4) {
            const unsigned long long ga =
                (unsigned long long)(uintptr_t)(gM4 + idx);
            // Low 32 bits of the generic shared pointer == LDS byte offset.
            const unsigned la = (unsigned)(uintptr_t)(sM4 + idx);
            asm volatile("global_store_async_from_lds_b128 %0, %1, off"
                         :: "v"(ga), "v"(la) : "memory");
        }
    }
    if (tid < (BLK * 3) / 4) {
        const unsigned long long ga = (unsigned long long)(uintptr_t)(gL4 + tid);
        const unsigned la = (unsigned)(uintptr_t)(sL4 + tid);
        asm volatile("global_store_async_from_lds_b128 %0, %1, off"
                     :: "v"(ga), "v"(la) : "memory");
    }
    // Drain async LDS->global traffic before the wave retires / LDS is freed.
    asm volatile("s_wait_asynccnt 0x0" ::: "memory");
}

// Tail kernel: plain scalar path for the (normally zero) ragged remainder.
__global__ __launch_bounds__(BLK)
void so3_encode_tail(const float* __restrict__ z,
                     float* __restrict__ out_mean,
                     float* __restrict__ out_logvar,
                     int first, int total)
{
    const int item = first + blockIdx.x * BLK + threadIdx.x;
    if (item >= total) return;
    const int brow = item >> 4;
    const int g    = item & (NG - 1);
    const float* __restrict__ va = z + (size_t)brow * (6 * NG) + 3 * g;
    const float* __restrict__ vl = va + 3 * NG;
    float m[9], lo[3];
    so3_item(va[0], va[1], va[2], vl[0], vl[1], vl[2], m, lo);
    float* gM = out_mean   + (size_t)item * 9;
    float* gL = out_logvar + (size_t)item * 3;
    #pragma unroll
    for (int j = 0; j < 9; ++j) gM[j] = m[j];
    #pragma unroll
    for (int j = 0; j < 3; ++j) gL[j] = lo[j];
}

extern "C" void kernel_launch(void* const* d_in, const int* in_sizes, int n_in,
                              void* d_out, int out_size, void* d_ws, size_t ws_size,
                              hipStream_t stream) {
    (void)n_in; (void)out_size; (void)d_ws; (void)ws_size;
    const float* z = (const float*)d_in[0];
    const int B     = in_sizes[0] / (6 * NG);   // 262144
    const int total = B * NG;                   // 4,194,304 items

    float* out_mean   = (float*)d_out;                   // B*16*9 floats
    float* out_logvar = out_mean + (size_t)B * NG * 9;   // B*16*3 floats

    const int fullBlocks = total / BLK;                  // 16384
    const int tailItems  = total - fullBlocks * BLK;     // 0 for this shape

    if (fullBlocks > 0) {
        so3_encode_main<<<fullBlocks, BLK, 0, stream>>>(z, out_mean, out_logvar);
    }
    if (tailItems > 0) {
        so3_encode_tail<<<(tailItems + BLK - 1) / BLK, BLK, 0, stream>>>(
            z, out_mean, out_logvar, fullBlocks * BLK, total);
    }
}